// XorGlia_64939905515614
// MI455X (gfx1250) — compile-verified
//
#include <hip/hip_runtime.h>

// CDNA5 / gfx1250, wave32.
typedef __attribute__((ext_vector_type(2))) float v2f;
typedef __attribute__((ext_vector_type(8))) float v8f;

// Banded locally-connected layer: y[i] = sum_{j in [i+C-1, i+C+1] ∩ [0,IN)} W[i][j]*h[j]
// All indices compile-time after unroll -> weights indexed uniformly (s_load),
// activations stay in VGPRs.
template <int IN, int OUT, int C>
__device__ __forceinline__ void lcband(const float* __restrict__ W,
                                       const float* __restrict__ h,
                                       float* __restrict__ y) {
#pragma unroll
  for (int i = 0; i < OUT; ++i) {
    float acc = 0.0f;
#pragma unroll
    for (int d = -1; d <= 1; ++d) {
      const int j = i + C + d;
      if (0 <= j && j < IN) acc = __builtin_fmaf(W[i * IN + j], h[j], acc);
    }
    y[i] = acc;
  }
}

template <int N>
__device__ __forceinline__ void softmax_(float* h) {
  float m = h[0];
#pragma unroll
  for (int i = 1; i < N; ++i) m = fmaxf(m, h[i]);
  float s = 0.0f;
#pragma unroll
  for (int i = 0; i < N; ++i) {
    h[i] = __expf(h[i] - m);  // v_exp_f32
    s += h[i];
  }
  const float r = __builtin_amdgcn_rcpf(s);  // v_rcp_f32 (scalar normalizer)
#pragma unroll
  for (int i = 0; i < N; ++i) h[i] *= r;
}

__global__ __launch_bounds__(256) void glia_forward(
    const float* __restrict__ x,
    const float* __restrict__ W0,  const float* __restrict__ W1,
    const float* __restrict__ W2,  const float* __restrict__ W3,
    const float* __restrict__ W4,  const float* __restrict__ W5,
    const float* __restrict__ W6,  const float* __restrict__ W7,
    const float* __restrict__ W8,  const float* __restrict__ W9,
    const float* __restrict__ W10, const float* __restrict__ W11,
    const float* __restrict__ W12,
    float* __restrict__ out, int B) {
  const int gid = blockIdx.x * 256 + (int)threadIdx.x;
  const int row = gid < B ? gid : B - 1;  // clamp: EXEC stays all-ones for WMMA
  const int lane = threadIdx.x & 31;
  const bool lo = lane < 16;
  const int m = lane & 15;

  // ---- load this lane's row (coalesced b64) ----
  const float2 xr = ((const float2*)x)[row];
  const float x0 = xr.x, x1 = xr.y;

  // ---- Layer 0 (SPREAD 2->4) via V_WMMA_F32_16X16X4_F32 ----
  // Band mask (c=-1): nonzeros (0,0) (1,0) (1,1) (2,0) (2,1) (3,1).
  // A operand = masked W0 [M=out(16, rows>=4 zero), K=4 (K>=2 zero)].
  // A layout: lanes 0-15 -> M=lane, VGPR0=K0, VGPR1=K1; lanes 16-31 -> K2,K3 (pad=0).
  const float w00 = W0[0], w10 = W0[2], w11 = W0[3];
  const float w20 = W0[4], w21 = W0[5], w31 = W0[7];
  const float a0 = lo ? ((m == 0) ? w00 : (m == 1) ? w10 : (m == 2) ? w20 : 0.0f) : 0.0f;
  const float a1 = lo ? ((m == 1) ? w11 : (m == 2) ? w21 : (m == 3) ? w31 : 0.0f) : 0.0f;
  const v2f A = {a0, a1};

  // B operand = x^T [K=4 (pad), N=16 batch cols].
  // B layout: N=lane%16; VGPR0 lower=K0 / upper=K2, VGPR1 lower=K1 / upper=K3.
  // Tile0 = rows of lanes 0-15; tile1 = rows of lanes 16-31 (half-swapped in).
  const float sx0 = __shfl_xor(x0, 16, 32);
  const float sx1 = __shfl_xor(x1, 16, 32);
  const v2f Bt0 = {lo ? x0 : 0.0f, lo ? x1 : 0.0f};
  const v2f Bt1 = {lo ? sx0 : 0.0f, lo ? sx1 : 0.0f};

  const v8f Cz = {};
  // (neg_a, A, neg_b, B, c_mod, C, reuse_a, reuse_b)
  const v8f D0 = __builtin_amdgcn_wmma_f32_16x16x4_f32(false, A, false, Bt0,
                                                       (short)0, Cz, false, false);
  const v8f D1 = __builtin_amdgcn_wmma_f32_16x16x4_f32(false, A, false, Bt1,
                                                       (short)0, Cz, false, false);

  // D layout: VGPR j, lanes 0-15 -> (M=j, N=lane). Merge the two tiles so every
  // lane holds its own row's 4 outputs: lane n>=16 takes tile1 col (n-16).
  float h4a[4];
#pragma unroll
  for (int j = 0; j < 4; ++j) {
    const float u = __shfl_xor(D1[j], 16, 32);
    h4a[j] = lo ? D0[j] : u;
  }
  softmax_<4>(h4a);

  // ---- Layers 1..12: band-aware per-lane VALU (optimal for 3-tap bands) ----
  float h4b[4];  lcband<4, 4, 0>(W1, h4a, h4b);   softmax_<4>(h4b);
  float h6a[6];  lcband<4, 6, -1>(W2, h4b, h6a);  softmax_<6>(h6a);
  float h6b[6];  lcband<6, 6, 0>(W3, h6a, h6b);   softmax_<6>(h6b);
  float h8a[8];  lcband<6, 8, -1>(W4, h6b, h8a);  softmax_<8>(h8a);
  float h8b[8];  lcband<8, 8, 0>(W5, h8a, h8b);   softmax_<8>(h8b);
  float hAa[10]; lcband<8, 10, -1>(W6, h8b, hAa); softmax_<10>(hAa);
  float hAb[10]; lcband<10, 10, 0>(W7, hAa, hAb); softmax_<10>(hAb);
  float g8[8];   lcband<10, 8, 1>(W8, hAb, g8);   softmax_<8>(g8);
  float g6[6];   lcband<8, 6, 1>(W9, g8, g6);     softmax_<6>(g6);
  float g4[4];   lcband<6, 4, 1>(W10, g6, g4);    softmax_<4>(g4);
  float g2[2];   lcband<4, 2, 1>(W11, g4, g2);    softmax_<2>(g2);

  // Final GATHER(2)->1, c=+1: y = W12[0]*g2[0] + W12[1]*g2[1], no softmax.
  const float y = __builtin_fmaf(W12[0], g2[0], W12[1] * g2[1]);

  if (gid < B) out[gid] = y;  // divergence after all WMMAs: safe
}

extern "C" void kernel_launch(void* const* d_in, const int* in_sizes, int n_in,
                              void* d_out, int out_size, void* d_ws, size_t ws_size,
                              hipStream_t stream) {
  (void)n_in; (void)out_size; (void)d_ws; (void)ws_size;
  const float* x = (const float*)d_in[0];
  const int B = in_sizes[0] / 2;  // x is [B, 2] float32
  const int blocks = (B + 255) / 256;
  glia_forward<<<blocks, 256, 0, stream>>>(
      x,
      (const float*)d_in[1],  (const float*)d_in[2],  (const float*)d_in[3],
      (const float*)d_in[4],  (const float*)d_in[5],  (const float*)d_in[6],
      (const float*)d_in[7],  (const float*)d_in[8],  (const float*)d_in[9],
      (const float*)d_in[10], (const float*)d_in[11], (const float*)d_in[12],
      (const float*)d_in[13],
      (float*)d_out, B);
}